// GaussianQuantRegularizer_6992206758164
// MI455X (gfx1250) — compile-verified
//
#include <hip/hip_runtime.h>

typedef __attribute__((ext_vector_type(2))) float v2f;
typedef __attribute__((ext_vector_type(8))) float v8f;

constexpr int   kB     = 4;
constexpr int   kL     = 1024;
constexpr int   kC2    = 512;
constexpr int   kC     = 256;    // latent channels per mu/logvar
constexpr int   kCG    = 64;     // C / GROUP
constexpr int   kRows  = kB * kL;        // 4096
constexpr int   kNQ    = kRows * kCG;    // 262144 queries
constexpr int   kNS    = 1024;           // prior samples
constexpr float kLog2Pi = 1.8378770664093453f;
constexpr float kLvMin  = -30.0f;
constexpr float kLvMax  = 20.0f;

// ---------------------------------------------------------------------------
// Kernel 1: zhat_noquant = mu + eps * exp(0.5 * clip(logvar))
// Pure streaming kernel, 1M elements, exact grid (no tail).
// ---------------------------------------------------------------------------
__global__ __launch_bounds__(256) void vae_noquant(const float* __restrict__ z,
                                                   const float* __restrict__ eps,
                                                   float* __restrict__ out) {
  const int i   = blockIdx.x * 256 + threadIdx.x;  // [0, kRows*kC)
  const int row = i >> 8;                          // / 256
  const int c   = i & 255;
  float mu = z[row * kC2 + c];
  float lv = z[row * kC2 + kC + c];
  lv = fminf(fmaxf(lv, kLvMin), kLvMax);
  out[i] = fmaf(eps[i], __expf(0.5f * lv), mu);
}

// ---------------------------------------------------------------------------
// Kernel 2: fused score-GEMM (K=8) + bias + row-argmax + codebook gather.
// One wave32 handles one 16-query tile; loops over 64 sample tiles with
// 2x V_WMMA_F32_16X16X4_F32 each (S features, S^2 features), bias in acc init.
// ---------------------------------------------------------------------------
__global__ __launch_bounds__(256) void vae_quant_wmma(const float* __restrict__ z,
                                                      const float* __restrict__ prior,   // [kNS, 4]
                                                      float* __restrict__ zhat,          // [kRows, kC]
                                                      float* __restrict__ idxf) {        // [kNQ]
  __shared__ float4 sp[kNS];      // 16 KB: prior rows
  __shared__ float  sbias[kNS];   // 4 KB : -beta * nlp_sum[n] = 0.5*sum(S^2) + 2*log2pi

  const int tid = threadIdx.x;
  for (int n = tid; n < kNS; n += 256) {
    float4 p = reinterpret_cast<const float4*>(prior)[n];
    sp[n] = p;
    float s2 = p.x * p.x + p.y * p.y + p.z * p.z + p.w * p.w;
    sbias[n] = 0.5f * s2 + 2.0f * kLog2Pi;
  }
  __syncthreads();

  const int  lane = tid & 31;
  const int  wave = tid >> 5;
  const int  tile = blockIdx.x * 8 + wave;   // 16-query tile id, grid covers exactly kNQ/16
  const int  m    = lane & 15;
  const bool hi   = lane >= 16;
  const int  g0   = hi ? 2 : 0;

  // ---- A fragments (16x4 f32 layout: lanes 0-15 hold K=0,1; lanes 16-31 K=2,3) ----
  const int k   = tile * 16 + m;   // query index
  const int row = k >> 6;          // b*L + l
  const int j   = k & 63;
  const int zb  = row * kC2 + j;

  float mu0 = z[zb + g0 * kCG];
  float mu1 = z[zb + (g0 + 1) * kCG];
  float lv0 = z[zb + kC + g0 * kCG];
  float lv1 = z[zb + kC + (g0 + 1) * kCG];
  lv0 = fminf(fmaxf(lv0, kLvMin), kLvMax);
  lv1 = fminf(fmaxf(lv1, kLvMin), kLvMax);
  const float iv0 = __expf(-lv0);
  const float iv1 = __expf(-lv1);

  v2f aS;  aS.x = mu0 * iv0;    aS.y = mu1 * iv1;     // features: mu * invvar
  v2f aQ;  aQ.x = -0.5f * iv0;  aQ.y = -0.5f * iv1;   // features: -0.5 * invvar

  float bestV[8];
  int   bestN[8];
#pragma unroll
  for (int v = 0; v < 8; ++v) { bestV[v] = -3.402823466e+38f; bestN[v] = 0; }

  for (int t = 0; t < kNS / 16; ++t) {
    const int n = t * 16 + m;          // sample column this lane owns (N = lane&15)
    float4 p = sp[n];
    v2f bS;                             // B 4x16 layout: lanes 0-15 K=0,1; lanes 16-31 K=2,3
    bS.x = hi ? p.z : p.x;
    bS.y = hi ? p.w : p.y;
    v2f bQ = bS * bS;
    const float bias = sbias[n];

    v8f acc;
#pragma unroll
    for (int v = 0; v < 8; ++v) acc[v] = bias;

    acc = __builtin_amdgcn_wmma_f32_16x16x4_f32(false, aS, false, bS, (short)0, acc, false, false);
    acc = __builtin_amdgcn_wmma_f32_16x16x4_f32(false, aQ, false, bQ, (short)0, acc, false, false);

    // Running argmax: VGPR v holds row M = v + 8*hi, column n. n increases with
    // t, so ">" preserves first-occurrence tie-breaking within a lane.
#pragma unroll
    for (int v = 0; v < 8; ++v) {
      float s = acc[v];
      if (s > bestV[v]) { bestV[v] = s; bestN[v] = n; }
    }
  }

  // Cross-lane argmax within each 16-lane half (xor 1,2,4,8 stays in-half on wave32).
#pragma unroll
  for (int v = 0; v < 8; ++v) {
    float bv = bestV[v];
    int   bn = bestN[v];
#pragma unroll
    for (int s = 1; s <= 8; s <<= 1) {
      float ov = __shfl_xor(bv, s, 32);
      int   on = __shfl_xor(bn, s, 32);
      if (ov > bv || (ov == bv && on < bn)) { bv = ov; bn = on; }
    }
    if (m == 0) {                         // lane 0 -> rows 0..7, lane 16 -> rows 8..15
      const int qm = v + (hi ? 8 : 0);
      const int q  = tile * 16 + qm;      // query index
      const int qr = q >> 6;
      const int qj = q & 63;
      idxf[q] = (float)bn;
      float4 pb = sp[bn];
      float* zr = zhat + qr * kC + qj;    // zhat[row, g*64 + j] = prior[idx][g]
      zr[0 * kCG] = pb.x;
      zr[1 * kCG] = pb.y;
      zr[2 * kCG] = pb.z;
      zr[3 * kCG] = pb.w;
    }
  }
}

// ---------------------------------------------------------------------------
extern "C" void kernel_launch(void* const* d_in, const int* in_sizes, int n_in,
                              void* d_out, int out_size, void* d_ws, size_t ws_size,
                              hipStream_t stream) {
  (void)in_sizes; (void)n_in; (void)out_size; (void)d_ws; (void)ws_size;
  const float* z     = (const float*)d_in[0];   // [4,1024,512]
  const float* eps   = (const float*)d_in[1];   // [4,1024,256]
  const float* prior = (const float*)d_in[2];   // [1024,4]

  float* out  = (float*)d_out;
  float* zhat = out;                       // [4,1024,256]
  float* noq  = out + kRows * kC;          // [4,1024,256]
  float* idxf = out + 2 * kRows * kC;      // [4,1024,64] (indices, stored as float)

  vae_noquant<<<(kRows * kC) / 256, 256, 0, stream>>>(z, eps, noq);
  vae_quant_wmma<<<(kNQ / 16) / 8, 256, 0, stream>>>(z, prior, zhat, idxf);
}